// NormalityModel_37658273252101
// MI455X (gfx1250) — compile-verified
//
#include <hip/hip_runtime.h>
#include <math.h>

typedef __attribute__((ext_vector_type(2))) float v2f;
typedef __attribute__((ext_vector_type(8))) float v8f;

// D = A(16x4) * B(4x16) + C, fp32, wave32
#define WMMA_F32_16X16X4(a, b, c) \
  __builtin_amdgcn_wmma_f32_16x16x4_f32(false, (a), false, (b), (short)0, (c), false, false)

constexpr int BN = 2048;            // batch rows
constexpr int DN = 2048;            // feature dim
constexpr int CN = 16;              // pca comps
constexpr int NN = 40000;           // kde points
constexpr int TILES  = NN / 16;     // 2500 dataset tiles (exact)
constexpr int NCHUNK = 50;          // N-chunks per row tile
constexpr int TPC    = TILES / NCHUNK; // 50 tiles per chunk (exact)

// ---------------------------------------------------------------------------
// Kernel 1: one wave per 16-row tile.
//   x = (F - mean) @ comps  via WMMA f32 16x16x4 over K=2048
//   y = (x / max_length) @ bw  via 4 more WMMAs (x staged through LDS)
//   also emits ynorm[m] = sum_c y[m][c]^2
// ---------------------------------------------------------------------------
__global__ __launch_bounds__(32) void pca_bw_kernel(
    const float* __restrict__ F, const float* __restrict__ mean,
    const float* __restrict__ comp, const float* __restrict__ maxlen,
    const float* __restrict__ bw, float* __restrict__ y,
    float* __restrict__ ynorm) {
  const int rowTile = blockIdx.x;        // 0..127
  const int lane = threadIdx.x;          // 0..31
  const int half = lane >> 4;            // 0 or 1
  const int l = lane & 15;

  const float* fr = F + (size_t)(rowTile * 16 + l) * DN;

  v8f acc = {0.f, 0.f, 0.f, 0.f, 0.f, 0.f, 0.f, 0.f};
  for (int k0 = 0; k0 < DN; k0 += 4) {
    const int ka = k0 + 2 * half;
    v2f a = *(const v2f*)(fr + ka);            // A: M=l, K={ka,ka+1}
    a.x -= mean[ka];
    a.y -= mean[ka + 1];
    v2f b;                                     // B: K={ka,ka+1}, N=l
    b.x = comp[(size_t)ka * CN + l];
    b.y = comp[(size_t)(ka + 1) * CN + l];
    acc = WMMA_F32_16X16X4(a, b, acc);
  }

  const float inv = 1.0f / maxlen[0];
  __shared__ float xs[16][16];
  #pragma unroll
  for (int r = 0; r < 8; ++r) xs[r + 8 * half][l] = acc[r] * inv;
  __syncthreads();

  v8f acc2 = {0.f, 0.f, 0.f, 0.f, 0.f, 0.f, 0.f, 0.f};
  #pragma unroll
  for (int ks = 0; ks < 4; ++ks) {
    const int ka = 4 * ks + 2 * half;
    v2f a2;
    a2.x = xs[l][ka];
    a2.y = xs[l][ka + 1];
    v2f b2;
    b2.x = bw[ka * CN + l];
    b2.y = bw[(ka + 1) * CN + l];
    acc2 = WMMA_F32_16X16X4(a2, b2, acc2);
  }

  float s[8];
  #pragma unroll
  for (int r = 0; r < 8; ++r) {
    const int m = r + 8 * half;                 // D layout: M = r + 8*half, N = l
    y[(size_t)(rowTile * 16 + m) * CN + l] = acc2[r];
    s[r] = acc2[r] * acc2[r];
  }
  #pragma unroll
  for (int off = 8; off >= 1; off >>= 1) {
    #pragma unroll
    for (int r = 0; r < 8; ++r) s[r] += __shfl_xor(s[r], off, 32);
  }
  if (l == 0) {
    #pragma unroll
    for (int r = 0; r < 8; ++r) ynorm[rowTile * 16 + r + 8 * half] = s[r];
  }
}

// ---------------------------------------------------------------------------
// Kernel 2: dnorm[n] = ||kde_dataset[n]||^2
// ---------------------------------------------------------------------------
__global__ void dnorm_kernel(const float* __restrict__ d,
                             float* __restrict__ dnorm) {
  const int i = blockIdx.x * blockDim.x + threadIdx.x;
  if (i < NN) {
    const float4* p = (const float4*)(d + (size_t)i * CN);
    float s = 0.f;
    #pragma unroll
    for (int j = 0; j < 4; ++j) {
      float4 v = p[j];
      s += v.x * v.x + v.y * v.y + v.z * v.z + v.w * v.w;
    }
    dnorm[i] = s;
  }
}

// ---------------------------------------------------------------------------
// Kernel 3: fused distance GEMM + exp + row reduction.
// One wave per (16-row tile, N-chunk). Per dataset tile: 4x WMMA f32 16x16x4
// computes S = Y_tile @ D_tileT; then sum_n exp(-0.5*(||y||^2+||d||^2-2S))
// is accumulated in registers (exp(-0.5*d2) = exp2(log2e*S - 0.5*log2e*(yn+dn))).
// ---------------------------------------------------------------------------
__global__ __launch_bounds__(32) void kde_kernel(
    const float* __restrict__ y, const float* __restrict__ ynorm,
    const float* __restrict__ dset, const float* __restrict__ dnorm,
    float* __restrict__ partial) {
  const int rowTile = blockIdx.x;        // 0..127
  const int chunk = blockIdx.y;          // 0..NCHUNK-1
  const int lane = threadIdx.x;
  const int half = lane >> 4;
  const int l = lane & 15;
  const float C1 = 1.44269504088896340736f;   // log2(e)

  // A operand (this 16x16 y tile) is invariant across the whole chunk.
  const float* yt = y + (size_t)rowTile * 16 * CN;
  v2f a[4];
  #pragma unroll
  for (int ks = 0; ks < 4; ++ks)
    a[ks] = *(const v2f*)(yt + l * CN + 4 * ks + 2 * half);

  float ynl[8];
  #pragma unroll
  for (int r = 0; r < 8; ++r)
    ynl[r] = ynorm[rowTile * 16 + r + 8 * half] * (-0.5f * C1);

  float sums[8] = {0.f, 0.f, 0.f, 0.f, 0.f, 0.f, 0.f, 0.f};

  const int t0 = chunk * TPC;
  for (int t = t0; t < t0 + TPC; ++t) {
    const float* dt = dset + (size_t)t * 16 * CN;   // 16x16 tile, row-major
    if (t + 1 < TILES)                              // prefetch next tile (1KB)
      __builtin_prefetch(dt + 256 + lane * 8, 0, 1);

    v8f acc = {0.f, 0.f, 0.f, 0.f, 0.f, 0.f, 0.f, 0.f};
    #pragma unroll
    for (int ks = 0; ks < 4; ++ks) {
      // B = D_tile^T : B[k][n] = dset_tile[n][k]; same addressing as A.
      v2f b = *(const v2f*)(dt + l * CN + 4 * ks + 2 * half);
      acc = WMMA_F32_16X16X4(a[ks], b, acc);
    }

    const float dnl = dnorm[t * 16 + l] * (-0.5f * C1);
    #pragma unroll
    for (int r = 0; r < 8; ++r)
      sums[r] += exp2f(fmaf(acc[r], C1, ynl[r] + dnl));
  }

  #pragma unroll
  for (int off = 8; off >= 1; off >>= 1) {
    #pragma unroll
    for (int r = 0; r < 8; ++r) sums[r] += __shfl_xor(sums[r], off, 32);
  }
  if (l == 0) {
    #pragma unroll
    for (int r = 0; r < 8; ++r)
      partial[(size_t)(rowTile * 16 + r + 8 * half) * NCHUNK + chunk] = sums[r];
  }
}

// ---------------------------------------------------------------------------
// Kernel 4: reduce chunk partials, log + logistic.
// ---------------------------------------------------------------------------
__global__ void final_kernel(const float* __restrict__ partial,
                             const float* __restrict__ kde_norm,
                             float* __restrict__ out) {
  const int i = blockIdx.x * blockDim.x + threadIdx.x;
  if (i < BN) {
    float s = 0.f;
    for (int c = 0; c < NCHUNK; ++c) s += partial[(size_t)i * NCHUNK + c];
    const float kde = s * (kde_norm[0] / (float)NN);  // mean * norm
    const float dens = logf(kde);                     // log(0) = -inf -> out 1
    out[i] = 1.0f / (1.0f + expf(0.05f * (dens - 12.0f)));
  }
}

extern "C" void kernel_launch(void* const* d_in, const int* in_sizes, int n_in,
                              void* d_out, int out_size, void* d_ws,
                              size_t ws_size, hipStream_t stream) {
  const float* features = (const float*)d_in[0];   // [B, D]
  const float* pca_mean = (const float*)d_in[1];   // [D]
  const float* pca_comp = (const float*)d_in[2];   // [D, C]
  const float* max_len  = (const float*)d_in[3];   // [1]
  const float* bw       = (const float*)d_in[4];   // [C, C]
  const float* dset     = (const float*)d_in[5];   // [N, C]
  const float* kde_norm = (const float*)d_in[6];   // [1]

  float* ws = (float*)d_ws;
  float* y       = ws;                         // 2048*16 = 32768
  float* ynorm   = ws + 32768;                 // 2048
  float* dnorm   = ws + 32768 + 2048;          // 40000
  float* partial = ws + 32768 + 2048 + 40000;  // 2048*NCHUNK = 102400

  pca_bw_kernel<<<dim3(BN / 16), dim3(32), 0, stream>>>(
      features, pca_mean, pca_comp, max_len, bw, y, ynorm);
  dnorm_kernel<<<dim3((NN + 255) / 256), dim3(256), 0, stream>>>(dset, dnorm);
  kde_kernel<<<dim3(BN / 16, NCHUNK), dim3(32), 0, stream>>>(
      y, ynorm, dset, dnorm, partial);
  final_kernel<<<dim3((BN + 255) / 256), dim3(256), 0, stream>>>(
      partial, kde_norm, (float*)d_out);
}